// Net_47450798686653
// MI455X (gfx1250) — compile-verified
//
#include <hip/hip_runtime.h>
#include <hip/hip_bf16.h>
#include <math.h>

typedef float v2f __attribute__((ext_vector_type(2)));
typedef float v8f __attribute__((ext_vector_type(8)));

#define HIDDEN 16

// ---------------- degree / normalization ----------------

__global__ void deg_init_kernel(float* __restrict__ deg, int n) {
    int i = blockIdx.x * blockDim.x + threadIdx.x;
    if (i < n) deg[i] = 1.0f;  // self-loop contributes 1 to every target degree
}

__global__ void deg_accum_kernel(const int* __restrict__ col, float* __restrict__ deg,
                                 int n_edges) {
    int e = blockIdx.x * blockDim.x + threadIdx.x;
    if (e < n_edges) atomicAdd(&deg[col[e]], 1.0f);
}

__global__ void deg_to_dinv_kernel(float* __restrict__ deg, int n) {
    int i = blockIdx.x * blockDim.x + threadIdx.x;
    if (i < n) {
        float d = deg[i];
        deg[i] = (d > 0.0f) ? rsqrtf(d) : 0.0f;
    }
}

// ---------------- layer-1 GEMM: h0 = x @ W1 (fp32 WMMA 16x16x4) ----------------
// One wave per 16-row tile. N = HIDDEN = 16 exactly.
// A 16x4 layout: lanes 0-15 rows M=0..15; VGPR0 = K{0|2}, VGPR1 = K{1|3} (hi half-wave +2).
// B 4x16 layout: lanes 0-15 cols N=0..15; same K striping.
// C/D 16x16: VGPR j -> row M = j + 8*(lane>=16), col N = lane&15.

__global__ void gemm1_wmma_kernel(const float* __restrict__ x, const float* __restrict__ W1,
                                  float* __restrict__ h0, int n, int nfeat) {
    int gtid = blockIdx.x * blockDim.x + threadIdx.x;
    int wave = gtid >> 5;
    int lane = threadIdx.x & 31;
    int m0 = wave * 16;
    if (m0 >= n) return;                       // wave-uniform: EXEC stays all-ones for WMMA
    int hi = lane >> 4;                        // 0 | 1
    int lr = lane & 15;

    int arow = m0 + lr;
    if (arow >= n) arow = n - 1;               // clamp (row m only feeds output row m)
    const float* A = x + (size_t)arow * nfeat;

    v8f c = {0.f, 0.f, 0.f, 0.f, 0.f, 0.f, 0.f, 0.f};
    for (int k = 0; k < nfeat; k += 4) {
        int kk = k + 2 * hi;
        v2f a;
        a.x = A[kk];
        a.y = A[kk + 1];
        v2f b;
        b.x = W1[kk * HIDDEN + lr];
        b.y = W1[(kk + 1) * HIDDEN + lr];
        c = __builtin_amdgcn_wmma_f32_16x16x4_f32(false, a, false, b, (short)0, c,
                                                  false, false);
    }
#pragma unroll
    for (int j = 0; j < 8; ++j) {
        int r = m0 + j + 8 * hi;
        if (r < n) h0[(size_t)r * HIDDEN + lr] = c[j];
    }
}

// ---------------- layer-2 GEMM: h1 = relu(agg1 + b1) @ W2 (fp32 WMMA) ----------------

__global__ void gemm2_wmma_kernel(const float* __restrict__ agg1, const float* __restrict__ b1,
                                  const float* __restrict__ W2, float* __restrict__ h1, int n) {
    int gtid = blockIdx.x * blockDim.x + threadIdx.x;
    int wave = gtid >> 5;
    int lane = threadIdx.x & 31;
    int m0 = wave * 16;
    if (m0 >= n) return;
    int hi = lane >> 4;
    int lr = lane & 15;

    int arow = m0 + lr;
    if (arow >= n) arow = n - 1;
    const float* A = agg1 + (size_t)arow * HIDDEN;

    v8f c = {0.f, 0.f, 0.f, 0.f, 0.f, 0.f, 0.f, 0.f};
#pragma unroll
    for (int k = 0; k < HIDDEN; k += 4) {
        int kk = k + 2 * hi;
        v2f a;
        a.x = fmaxf(A[kk] + b1[kk], 0.0f);         // fused bias + ReLU
        a.y = fmaxf(A[kk + 1] + b1[kk + 1], 0.0f);
        v2f b;
        b.x = W2[kk * HIDDEN + lr];
        b.y = W2[(kk + 1) * HIDDEN + lr];
        c = __builtin_amdgcn_wmma_f32_16x16x4_f32(false, a, false, b, (short)0, c,
                                                  false, false);
    }
#pragma unroll
    for (int j = 0; j < 8; ++j) {
        int r = m0 + j + 8 * hi;
        if (r < n) h1[(size_t)r * HIDDEN + lr] = c[j];
    }
}

// ---------------- scatter-add aggregation ----------------
// Self-loop term initializes every output element: out[i] = feat[i] * dinv[i]^2.

__global__ void init_self_kernel(const float* __restrict__ feat, const float* __restrict__ dinv,
                                 float* __restrict__ out, int n) {
    long long t = (long long)blockIdx.x * blockDim.x + threadIdx.x;
    if (t >= (long long)n * HIDDEN) return;
    int i = (int)(t >> 4);
    float d = dinv[i];
    out[t] = feat[t] * d * d;
}

// 16 consecutive lanes own one edge's 16 features -> coalesced gather + coalesced atomics.
__global__ void scatter_kernel(const float* __restrict__ feat, const int* __restrict__ row,
                               const int* __restrict__ col, const float* __restrict__ dinv,
                               float* __restrict__ out, int n_edges) {
    long long t = (long long)blockIdx.x * blockDim.x + threadIdx.x;
    long long e = t >> 4;
    int f = (int)(t & 15);
    if (e >= n_edges) return;
    int r = row[e];
    int cidx = col[e];
    float nrm = dinv[r] * dinv[cidx];
    float v = feat[(size_t)r * HIDDEN + f] * nrm;
    atomicAdd(&out[(size_t)cidx * HIDDEN + f], v);
}

// ---------------- bias + log_softmax over 16 classes (16-lane shuffle reduce) ----------------

__global__ void logsoftmax_kernel(float* __restrict__ out, const float* __restrict__ b2, int n) {
    long long t = (long long)blockIdx.x * blockDim.x + threadIdx.x;
    if (t >= (long long)n * HIDDEN) return;
    int f = (int)(t & 15);
    float v = out[t] + b2[f];

    float m = v;
#pragma unroll
    for (int off = 1; off < 16; off <<= 1) m = fmaxf(m, __shfl_xor(m, off, 16));
    float ex = expf(v - m);
    float s = ex;
#pragma unroll
    for (int off = 1; off < 16; off <<= 1) s += __shfl_xor(s, off, 16);

    out[t] = v - m - logf(s);
}

// ---------------- host-side launch ----------------

extern "C" void kernel_launch(void* const* d_in, const int* in_sizes, int n_in,
                              void* d_out, int out_size, void* d_ws, size_t ws_size,
                              hipStream_t stream) {
    (void)n_in; (void)out_size; (void)ws_size;

    const float* x  = (const float*)d_in[0];
    const int*   ei = (const int*)d_in[1];
    const float* W1 = (const float*)d_in[2];
    const float* b1 = (const float*)d_in[3];
    const float* W2 = (const float*)d_in[4];
    const float* b2 = (const float*)d_in[5];
    float* out = (float*)d_out;

    const int nfeat = in_sizes[2] / HIDDEN;   // 256
    const int n = in_sizes[0] / nfeat;        // 100000
    const int E = in_sizes[1] / 2;            // 3200000
    const int* row = ei;      // sources
    const int* col = ei + E;  // targets

    // workspace layout (floats): dinv[n] | bufA[n*16] | bufB[n*16]
    float* dinv = (float*)d_ws;
    float* bufA = dinv + n;                      // h0, later reused for h1
    float* bufB = bufA + (size_t)n * HIDDEN;     // layer-1 aggregation

    const int TB = 256;
    // 1) symmetric normalization: deg = 1 + indegree; dinv = rsqrt(deg)
    deg_init_kernel<<<(n + TB - 1) / TB, TB, 0, stream>>>(dinv, n);
    deg_accum_kernel<<<(E + TB - 1) / TB, TB, 0, stream>>>(col, dinv, E);
    deg_to_dinv_kernel<<<(n + TB - 1) / TB, TB, 0, stream>>>(dinv, n);

    // 2) h0 = x @ W1 (fp32 WMMA)
    int tiles = (n + 15) / 16;
    long long gthreads = (long long)tiles * 32;
    int gblocks = (int)((gthreads + 127) / 128);
    gemm1_wmma_kernel<<<gblocks, 128, 0, stream>>>(x, W1, bufA, n, nfeat);

    // 3) layer-1 aggregation: self-loop init + edge scatter
    long long nt = (long long)n * HIDDEN;
    init_self_kernel<<<(int)((nt + TB - 1) / TB), TB, 0, stream>>>(bufA, dinv, bufB, n);
    long long st = (long long)E * HIDDEN;
    scatter_kernel<<<(int)((st + TB - 1) / TB), TB, 0, stream>>>(bufA, row, col, dinv, bufB, E);

    // 4) h1 = relu(agg1 + b1) @ W2 (fp32 WMMA, bias+ReLU fused into A load)
    gemm2_wmma_kernel<<<gblocks, 128, 0, stream>>>(bufB, b1, W2, bufA, n);

    // 5) layer-2 aggregation straight into d_out
    init_self_kernel<<<(int)((nt + TB - 1) / TB), TB, 0, stream>>>(bufA, dinv, out, n);
    scatter_kernel<<<(int)((st + TB - 1) / TB), TB, 0, stream>>>(bufA, row, col, dinv, out, E);

    // 6) + b2, row-wise log_softmax (in place)
    logsoftmax_kernel<<<(int)((nt + TB - 1) / TB), TB, 0, stream>>>(out, b2, n);
}